// InHarmonic_62929860821362
// MI455X (gfx1250) — compile-verified
//
#include <hip/hip_runtime.h>
#include <cstdint>

// ---------------- problem constants (from reference) ----------------
#define BATCH     8
#define T_FRAMES  750
#define N_HARM    64
#define UPS       96
#define N_SAMP    (T_FRAMES * UPS)      // 72000
#define NYQ       12000.0f              // SAMPLE_RATE / 2
#define INV_SR    (1.0f / 24000.0f)
#define LN10      2.302585092994046f
#define LOG2E     1.4426950408889634f

typedef __attribute__((ext_vector_type(2))) float v2f;
typedef __attribute__((ext_vector_type(8))) float v8f;

// DDSP exp_sigmoid: 2 * sigmoid(x)^ln(10) + 1e-7
__device__ __forceinline__ float scale_fn(float x) {
    float e = __builtin_amdgcn_exp2f(-x * LOG2E);          // exp(-x)
    float s = __builtin_amdgcn_rcpf(1.0f + e);             // sigmoid
    float p = __builtin_amdgcn_exp2f(LN10 * __builtin_amdgcn_logf(s)); // s^ln10
    return 2.0f * p + 1e-7f;
}

// ---------------------------------------------------------------
// Kernel 1: frame-rate controls.  One block per (b,t) frame, 64 threads
// (one per harmonic).  Produces freq_ws (B,T,H) and amp_ws (B,T,H).
// ---------------------------------------------------------------
__global__ void __launch_bounds__(N_HARM)
frame_kernel(const float* __restrict__ amp_in,   // (B,T,1)
             const float* __restrict__ hd_in,    // (B,T,H)
             const float* __restrict__ inh_in,   // (B,T,1)
             const float* __restrict__ f0_in,    // (B,T,1)
             float* __restrict__ freq_ws,
             float* __restrict__ amp_ws) {
    const int bt = blockIdx.x;           // 0 .. B*T-1
    const int h  = threadIdx.x;          // 0 .. 63

    const float f0 = f0_in[bt];
    const float Bc = fmaxf(inh_in[bt], 0.0f);
    const float a  = scale_fn(amp_in[bt]);

    const float kf = (float)(h + 1);
    float hd = scale_fn(hd_in[bt * N_HARM + h]);

    // inharmonic frequency: f0 * k * sqrt(1 + B k^2)
    // (reference's shift round-trip reproduces exactly this value)
    const float fr = f0 * kf * sqrtf(fmaf(Bc * kf, kf, 1.0f));

    hd = (fr < NYQ) ? hd : 0.0f;         // remove_above_nyquist at frame rate

    __shared__ float red[N_HARM];
    red[h] = hd;
    __syncthreads();
#pragma unroll
    for (int off = 32; off > 0; off >>= 1) {
        if (h < off) red[h] += red[h + off];
        __syncthreads();
    }
    float den = red[0];
    den = (den == 0.0f) ? 1e-7f : den;   // safe_divide

    const float am = (f0 > 20.0f) ? a : 0.0f;   // MIN_FREQUENCY gate

    const int idx = bt * N_HARM + h;
    freq_ws[idx] = fr;
    amp_ws[idx]  = am * hd * __builtin_amdgcn_rcpf(den);
}

// ---------------------------------------------------------------
// Kernel 2: per-(b,h) scan of segment phase sums (in CYCLES, not rad).
// Linear upsampling => segment i contributes inv_sr*(96*f_i + df*47.5)
// where df = f_{min(i+1,T-1)} - f_i  (sum_{j=0..95} j / 96 = 47.5).
// base_ws[b,i,h] = cumulative cycles before the first sample of segment i.
// 512 independent scans of length 750; wave-coalesced over h.
// ---------------------------------------------------------------
__global__ void __launch_bounds__(256)
phase_kernel(const float* __restrict__ freq_ws,
             float* __restrict__ base_ws) {
    const int t = blockIdx.x * blockDim.x + threadIdx.x;   // 0..511
    const int b = t >> 6;
    const int h = t & 63;
    const float* fp = freq_ws + (size_t)b * T_FRAMES * N_HARM + h;
    float*       bp = base_ws + (size_t)b * T_FRAMES * N_HARM + h;

    float acc = 0.0f;
    float fi  = fp[0];
    for (int i = 0; i < T_FRAMES; ++i) {
        bp[i * N_HARM] = acc;
        const float fn = (i + 1 < T_FRAMES) ? fp[(i + 1) * N_HARM] : fi;
        acc += INV_SR * fmaf(fn - fi, 47.5f, 96.0f * fi);
        fi = fn;
    }
}

// ---------------------------------------------------------------
// Kernel 3: sample-rate synthesis.  One wave32 per 16-sample tile.
// 96 % 16 == 0 => a tile never crosses a frame-segment boundary, so all
// lanes share i0/i1.  Lanes 0-15 carry samples m=0..15 (A-matrix rows);
// lane-half selects K pair per V_WMMA_F32_16X16X4_F32 layout:
//   VGPR0: K = 2*(lane/16),  VGPR1: K = 2*(lane/16)+1.
// Accumulating 16 WMMAs (K chunks of 4) against an all-ones B gives
// D[m][*] = sum over all 64 harmonics of cos(2*pi*cycles)*amp*mask.
// cos via v_cos_f32 (computes cos(2*pi*x)) -- phase carried in cycles.
// ---------------------------------------------------------------
__global__ void __launch_bounds__(256)
synth_kernel(const float* __restrict__ freq_ws,
             const float* __restrict__ amp_ws,
             const float* __restrict__ base_ws,
             float* __restrict__ out) {
    const int lane = threadIdx.x & 31;
    const int tile = blockIdx.x * 8 + (threadIdx.x >> 5);  // 0 .. 35999
    const int b    = tile / (N_SAMP / 16);                 // /4500
    const int ts   = (tile - b * (N_SAMP / 16)) * 16;      // tile start sample
    const int m    = lane & 15;                            // A-matrix row
    const int kh   = lane >> 4;                            // K half select
    const int s    = ts + m;
    const int i0   = s / UPS;                              // uniform in tile
    const int j    = s - i0 * UPS;                         // 0..95, per lane
    const int i1   = (i0 + 1 < T_FRAMES) ? i0 + 1 : i0;

    const int base0 = (b * T_FRAMES + i0) * N_HARM;
    const int base1 = (b * T_FRAMES + i1) * N_HARM;

    const float jp1 = (float)(j + 1);
    const float tri = (float)((j * (j + 1)) >> 1) * (1.0f / 96.0f); // sum_{j'<=j} j'/96
    const float w   = (float)j * (1.0f / 96.0f);                    // lerp weight

    v8f acc = {};                      // C/D accumulator (16x16 f32)
    v2f ones; ones.x = 1.0f; ones.y = 1.0f;   // B = all ones (4x16)

#pragma unroll 2
    for (int kb = 0; kb < N_HARM; kb += 4) {
        const int h = kb + (kh << 1);  // even => 8B aligned v2f loads
        const v2f f0p = *(const v2f*)(freq_ws + base0 + h);
        const v2f f1p = *(const v2f*)(freq_ws + base1 + h);
        const v2f a0p = *(const v2f*)(amp_ws  + base0 + h);
        const v2f a1p = *(const v2f*)(amp_ws  + base1 + h);
        const v2f pbp = *(const v2f*)(base_ws + base0 + h);

        v2f A;
        {   // harmonic h  -> A-matrix element (m, K=2*kh)
            const float df  = f1p.x - f0p.x;
            const float cyc = fmaf(INV_SR, fmaf(jp1, f0p.x, df * tri), pbp.x);
            const float fr  = fmaf(df, w, f0p.x);
            const float am  = fmaf(a1p.x - a0p.x, w, a0p.x);
            const float cv  = __builtin_amdgcn_cosf(cyc);  // cos(2*pi*cyc)
            A.x = (fr < NYQ) ? cv * am : 0.0f;
        }
        {   // harmonic h+1 -> A-matrix element (m, K=2*kh+1)
            const float df  = f1p.y - f0p.y;
            const float cyc = fmaf(INV_SR, fmaf(jp1, f0p.y, df * tri), pbp.y);
            const float fr  = fmaf(df, w, f0p.y);
            const float am  = fmaf(a1p.y - a0p.y, w, a0p.y);
            const float cv  = __builtin_amdgcn_cosf(cyc);
            A.y = (fr < NYQ) ? cv * am : 0.0f;
        }
        // D = A(16x4) x ones(4x16) + C : every column of D = row sum
        acc = __builtin_amdgcn_wmma_f32_16x16x4_f32(
                  false, A, false, ones, (short)0, acc, false, false);
    }

    // D layout: VGPR v, lanes 0-15 -> M=v, lanes 16-31 -> M=v+8 (any N).
    // Lanes with N==0 (lanes 0 and 16) write the 16 sample sums.
    if (m == 0) {
        float* o = out + (size_t)b * N_SAMP + ts + kh * 8;
#pragma unroll
        for (int v = 0; v < 8; ++v) o[v] = acc[v];
    }
}

// ---------------------------------------------------------------
extern "C" void kernel_launch(void* const* d_in, const int* in_sizes, int n_in,
                              void* d_out, int out_size, void* d_ws, size_t ws_size,
                              hipStream_t stream) {
    (void)in_sizes; (void)n_in; (void)out_size; (void)ws_size;
    const float* amp_in = (const float*)d_in[0];   // amplitudes (8,750,1)
    const float* hd_in  = (const float*)d_in[1];   // harmonic_distribution (8,750,64)
    const float* inh_in = (const float*)d_in[2];   // inharm_coef (8,750,1)
    const float* f0_in  = (const float*)d_in[3];   // f0_hz (8,750,1)
    float* out = (float*)d_out;                    // (8,72000)

    const size_t bth = (size_t)BATCH * T_FRAMES * N_HARM;   // 384000 floats
    float* freq_ws = (float*)d_ws;                 // (B,T,H)
    float* amp_ws  = freq_ws + bth;                // (B,T,H)
    float* base_ws = amp_ws  + bth;                // (B,T,H) cycles

    frame_kernel<<<BATCH * T_FRAMES, N_HARM, 0, stream>>>(
        amp_in, hd_in, inh_in, f0_in, freq_ws, amp_ws);
    phase_kernel<<<2, 256, 0, stream>>>(freq_ws, base_ws);
    synth_kernel<<<(BATCH * (N_SAMP / 16)) / 8, 256, 0, stream>>>(
        freq_ws, amp_ws, base_ws, out);
}